// GatingNetwork_79044578116182
// MI455X (gfx1250) — compile-verified
//
#include <hip/hip_runtime.h>
#include <hip/hip_bf16.h>

// Problem constants (from reference)
constexpr int N   = 128;
constexpr int M   = 256;
constexpr int E   = 64;
constexpr int S   = 1024;
constexpr int B   = 512;
constexpr int Sm1 = S - 1;       // 1023
constexpr int NM  = N + M;       // 384

typedef __attribute__((ext_vector_type(2))) float v2f;
typedef __attribute__((ext_vector_type(8))) float v8f;

// D = A(16x4 f32) * B(4x16 f32) + C(16x16 f32), one tile per wave32.
__device__ __forceinline__ v8f wmma4(v2f a, v2f b, v8f c) {
    return __builtin_amdgcn_wmma_f32_16x16x4_f32(
        /*neg_a=*/false, a, /*neg_b=*/false, b,
        /*c_mod=*/(short)0, c, /*reuse_a=*/false, /*reuse_b=*/false);
}

// CDNA5 async global->LDS copy path (guarded so both toolchains compile).
#if defined(__gfx1250__) && __has_builtin(__builtin_amdgcn_global_load_async_to_lds_b128) && \
    __has_builtin(__builtin_amdgcn_s_wait_asynccnt)
#define HAVE_ASYNC_LDS 1
typedef int v4i_gcc __attribute__((vector_size(16)));   // b128 payload type
__device__ __forceinline__ void async_copy16(const float* g, float* l) {
    __builtin_amdgcn_global_load_async_to_lds_b128(
        (__attribute__((address_space(1))) v4i_gcc*)(g),
        (__attribute__((address_space(3))) v4i_gcc*)(l), 0, 0);
}
#else
#define HAVE_ASYNC_LDS 0
#endif

// ---------------------------------------------------------------------------
// K1: z_current^T[b][n] = (D @ z)[n][b] + sigma[n]*noise[n][b]
// grid = B, block = N(=128)
__global__ void k_zcur(const float* __restrict__ z, const float* __restrict__ noise,
                       const float* __restrict__ Dm, const float* __restrict__ sigma,
                       float* __restrict__ zc_t) {
    __shared__ float zcol[M];
    const int b = blockIdx.x;
    const int n = threadIdx.x;           // 0..127
    zcol[n]     = z[n * B + b];
    zcol[n + N] = z[(n + N) * B + b];
    __syncthreads();
    float acc = sigma[n] * noise[n * B + b];
    #pragma unroll 8
    for (int m = 0; m < M; ++m) acc += Dm[n * M + m] * zcol[m];
    zc_t[b * N + n] = acc;
}

// ---------------------------------------------------------------------------
// K2 (fused, single context pass): per column b
//   d[s]   = sum_n |c[s,b,n] - zc[b,n]|
//   w[s]   = softmax_s(-d[s]/|temp1|)
//   u0[n]  = sum_s w[s] c[s,b,n] ;  u1[n] = sum_s w[s] c[s+1,b,n]
// Flash-style online softmax, single read of context (256 MB total).
// 8 waves per block; wave w owns contiguous frames [w*128, w*128+128)
// (+1 overlap frame for the u1 shift). Context is streamed through a 4-deep
// per-wave LDS ring via async global->LDS copies (prefetch distance 3,
// steady-state s_wait_asynccnt 3). Partial softmax states merged via LDS.
// grid = B, block = 256
__global__ void k_attn(const float* __restrict__ context, const float* __restrict__ zc_t,
                       const float* __restrict__ temp1,
                       float* __restrict__ u0, float* __restrict__ u1) {
    constexpr int WAVES = 8;
    constexpr int CHUNK = S / WAVES;          // 128 frames per wave
    __shared__ float sm[WAVES];               // per-wave running max
    __shared__ float sl[WAVES];               // per-wave running sum
    __shared__ __align__(16) float sa0[WAVES][N];
    __shared__ __align__(16) float sa1[WAVES][N];
#if HAVE_ASYNC_LDS
    __shared__ __align__(16) float abuf[WAVES][4][N];  // 4-deep ring per wave
#endif
    const int b    = blockIdx.x;
    const int lane = threadIdx.x & 31;
    const int w    = threadIdx.x >> 5;
    const float invT = 1.0f / fabsf(temp1[0]);

    const int lo    = w * CHUNK;
    const int hi    = lo + CHUNK;                    // this wave weights s in [lo,hi)
    const int fLast = (hi < S) ? hi : (S - 1);       // inclusive last frame to read

    const float4 zz = *(const float4*)(zc_t + (size_t)b * N + lane * 4);

    float m = -3.402823e38f, suml = 0.f, eprev = 0.f;
    float4 a0 = {0.f, 0.f, 0.f, 0.f};
    float4 a1 = {0.f, 0.f, 0.f, 0.f};

#if HAVE_ASYNC_LDS
    // prologue: 3 frames in flight
    #pragma unroll
    for (int q = 0; q < 3; ++q)
        async_copy16(context + ((size_t)(lo + q) * B + b) * N + lane * 4,
                     &abuf[w][q][lane * 4]);
#endif
    for (int f = lo; f <= fLast; ++f) {
        float4 c4;
#if HAVE_ASYNC_LDS
        const int p = (f - lo) & 3;
        if (f + 3 <= fLast) {
            async_copy16(context + ((size_t)(f + 3) * B + b) * N + lane * 4,
                         &abuf[w][(p + 3) & 3][lane * 4]);
            __builtin_amdgcn_s_wait_asynccnt(3);     // frame f's copy complete
        } else {
            __builtin_amdgcn_s_wait_asynccnt(0);     // tail: drain
        }
        c4 = *(const float4*)(&abuf[w][p][lane * 4]);
#else
        c4 = *(const float4*)(context + ((size_t)f * B + b) * N + lane * 4);
#endif
        const bool hasw = (f < hi) && (f < Sm1);
        float l = -3.402823e38f;
        if (hasw) {
            float pd = fabsf(c4.x - zz.x) + fabsf(c4.y - zz.y) +
                       fabsf(c4.z - zz.z) + fabsf(c4.w - zz.w);
            #pragma unroll
            for (int off = 16; off; off >>= 1) pd += __shfl_xor(pd, off, 32);
            l = -pd * invT;
        }
        const float newm = fmaxf(m, l);
        const float sc   = __expf(m - newm);
        const float e    = hasw ? __expf(l - newm) : 0.f;
        suml = suml * sc + e;
        a0.x = a0.x * sc + e * c4.x;  a0.y = a0.y * sc + e * c4.y;
        a0.z = a0.z * sc + e * c4.z;  a0.w = a0.w * sc + e * c4.w;
        a1.x = (a1.x + eprev * c4.x) * sc;  a1.y = (a1.y + eprev * c4.y) * sc;
        a1.z = (a1.z + eprev * c4.z) * sc;  a1.w = (a1.w + eprev * c4.w) * sc;
        eprev = e;
        m = newm;
    }

    if (lane == 0) { sm[w] = m; sl[w] = suml; }
    *(float4*)(&sa0[w][lane * 4]) = a0;
    *(float4*)(&sa1[w][lane * 4]) = a1;
    __syncthreads();

    // merge the 8 partial softmax states
    float gm = -3.402823e38f;
    #pragma unroll
    for (int i = 0; i < WAVES; ++i) gm = fmaxf(gm, sm[i]);
    float esc[WAVES], total = 0.f;
    #pragma unroll
    for (int i = 0; i < WAVES; ++i) { esc[i] = __expf(sm[i] - gm); total += esc[i] * sl[i]; }
    const float invtot = 1.0f / total;

    const int t = threadIdx.x;
    if (t < N) {
        float acc = 0.f;
        #pragma unroll
        for (int i = 0; i < WAVES; ++i) acc += sa0[i][t] * esc[i];
        u0[b * N + t] = acc * invtot;
    } else {
        const int n = t - N;
        float acc = 0.f;
        #pragma unroll
        for (int i = 0; i < WAVES; ++i) acc += sa1[i][n] * esc[i];
        u1[b * N + n] = acc * invtot;
    }
}

// ---------------------------------------------------------------------------
// K3 (WMMA): comb[b][n] = sum_i u0[b][i]*cw[n][i][0] + u1[b][i]*cw[n][i][1] + cb[n]
// one wave per 16x16 tile; grid = (B/16)*(N/16) = 256, block = 32
__global__ void k_embed(const float* __restrict__ u0, const float* __restrict__ u1,
                        const float* __restrict__ cw, const float* __restrict__ cb,
                        float* __restrict__ comb) {
    const int t  = blockIdx.x;
    const int bt = (t / (N / 16)) * 16;
    const int nt = (t % (N / 16)) * 16;
    const int lane = threadIdx.x;
    const int row  = lane & 15;          // A-row (b) and B-col (n) index
    const int kb   = (lane >> 4) << 1;   // 0 or 2
    v8f acc = {};
    for (int k0 = 0; k0 < N; k0 += 4) {
        const int k = k0 + kb;
        v2f a0 = *(const v2f*)(u0 + (bt + row) * N + k);
        v2f b0 = { cw[(nt + row) * (N * 2) + k * 2 + 0],
                   cw[(nt + row) * (N * 2) + (k + 1) * 2 + 0] };
        acc = wmma4(a0, b0, acc);
        v2f a1 = *(const v2f*)(u1 + (bt + row) * N + k);
        v2f b1 = { cw[(nt + row) * (N * 2) + k * 2 + 1],
                   cw[(nt + row) * (N * 2) + (k + 1) * 2 + 1] };
        acc = wmma4(a1, b1, acc);
    }
    const int half = lane >> 4, ccol = lane & 15;
    #pragma unroll
    for (int r = 0; r < 8; ++r) {
        const int crow = r + (half ? 8 : 0);
        comb[(bt + crow) * NM + nt + ccol] = acc[r] + cb[nt + ccol];
    }
}

// K3b: comb[b][N+m] = z[m][b]   (grid = B*M/256, block = 256)
__global__ void k_fillz(const float* __restrict__ z, float* __restrict__ comb) {
    const int idx = blockIdx.x * blockDim.x + threadIdx.x;
    const int b = idx / M, m = idx % M;
    comb[b * NM + N + m] = z[m * B + b];
}

// ---------------------------------------------------------------------------
// K4 (WMMA): h[b][e] = relu(sum_k comb[b][k]*W1[e][k] + b1[e])
// grid = (B/16)*(E/16) = 128, block = 32
__global__ void k_mlp1(const float* __restrict__ comb, const float* __restrict__ W1,
                       const float* __restrict__ b1, float* __restrict__ h) {
    const int t  = blockIdx.x;
    const int bt = (t / (E / 16)) * 16;
    const int et = (t % (E / 16)) * 16;
    const int lane = threadIdx.x;
    const int row  = lane & 15;
    const int kb   = (lane >> 4) << 1;
    v8f acc = {};
    for (int k0 = 0; k0 < NM; k0 += 4) {
        const int k = k0 + kb;
        v2f a = *(const v2f*)(comb + (bt + row) * NM + k);
        v2f w = *(const v2f*)(W1 + (et + row) * NM + k);
        acc = wmma4(a, w, acc);
    }
    const int half = lane >> 4, ccol = lane & 15;
    #pragma unroll
    for (int r = 0; r < 8; ++r) {
        const int crow = r + (half ? 8 : 0);
        h[(bt + crow) * E + et + ccol] = fmaxf(acc[r] + b1[et + ccol], 0.0f);
    }
}

// K5 (WMMA): mlp[b][e] = sum_k h[b][k]*W2[e][k] + b2[e]
// grid = 128, block = 32
__global__ void k_mlp2(const float* __restrict__ h, const float* __restrict__ W2,
                       const float* __restrict__ b2, float* __restrict__ mlp) {
    const int t  = blockIdx.x;
    const int bt = (t / (E / 16)) * 16;
    const int et = (t % (E / 16)) * 16;
    const int lane = threadIdx.x;
    const int row  = lane & 15;
    const int kb   = (lane >> 4) << 1;
    v8f acc = {};
    for (int k0 = 0; k0 < E; k0 += 4) {
        const int k = k0 + kb;
        v2f a = *(const v2f*)(h + (bt + row) * E + k);
        v2f w = *(const v2f*)(W2 + (et + row) * E + k);
        acc = wmma4(a, w, acc);
    }
    const int half = lane >> 4, ccol = lane & 15;
    #pragma unroll
    for (int r = 0; r < 8; ++r) {
        const int crow = r + (half ? 8 : 0);
        mlp[(bt + crow) * E + et + ccol] = acc[r] + b2[et + ccol];
    }
}

// ---------------------------------------------------------------------------
// K6: out[e][b] = softmax_e(-mlp[b][e]/|temp2|)   (one wave per b, 2 e's per lane)
// grid = B/8, block = 256
__global__ void k_softmax_e(const float* __restrict__ mlp, const float* __restrict__ temp2,
                            float* __restrict__ out) {
    const int lane = threadIdx.x & 31;
    const int b = blockIdx.x * 8 + (threadIdx.x >> 5);
    const float invT = 1.0f / fabsf(temp2[0]);
    float l0 = -mlp[b * E + lane] * invT;
    float l1 = -mlp[b * E + lane + 32] * invT;
    float m = fmaxf(l0, l1);
    #pragma unroll
    for (int off = 16; off; off >>= 1) m = fmaxf(m, __shfl_xor(m, off, 32));
    float e0 = __expf(l0 - m), e1 = __expf(l1 - m);
    float sum = e0 + e1;
    #pragma unroll
    for (int off = 16; off; off >>= 1) sum += __shfl_xor(sum, off, 32);
    const float inv = 1.0f / sum;
    out[lane * B + b]        = e0 * inv;
    out[(lane + 32) * B + b] = e1 * inv;
}

// ---------------------------------------------------------------------------
extern "C" void kernel_launch(void* const* d_in, const int* in_sizes, int n_in,
                              void* d_out, int out_size, void* d_ws, size_t ws_size,
                              hipStream_t stream) {
    const float* context = (const float*)d_in[0];
    const float* z       = (const float*)d_in[1];
    const float* noise   = (const float*)d_in[2];
    const float* conv_w  = (const float*)d_in[3];
    const float* conv_b  = (const float*)d_in[4];
    const float* Dm      = (const float*)d_in[5];
    const float* sigma   = (const float*)d_in[6];
    const float* temp1   = (const float*)d_in[7];
    const float* W1      = (const float*)d_in[8];
    const float* b1      = (const float*)d_in[9];
    const float* W2      = (const float*)d_in[10];
    const float* b2      = (const float*)d_in[11];
    const float* temp2   = (const float*)d_in[12];
    float* out = (float*)d_out;

    // workspace carve-up (~1.9 MB of f32)
    float* ws   = (float*)d_ws;
    float* zc_t = ws;  ws += B * N;               // [B][N]
    float* u0   = ws;  ws += B * N;               // [B][N]
    float* u1   = ws;  ws += B * N;               // [B][N]
    float* comb = ws;  ws += B * NM;              // [B][N+M]
    float* h    = ws;  ws += B * E;               // [B][E]
    float* mlp  = ws;  ws += B * E;               // [B][E]

    k_zcur<<<B, N, 0, stream>>>(z, noise, Dm, sigma, zc_t);
    k_attn<<<B, 256, 0, stream>>>(context, zc_t, temp1, u0, u1);
    k_embed<<<(B / 16) * (N / 16), 32, 0, stream>>>(u0, u1, conv_w, conv_b, comb);
    k_fillz<<<(B * M) / 256, 256, 0, stream>>>(z, comb);
    k_mlp1<<<(B / 16) * (E / 16), 32, 0, stream>>>(comb, W1, b1, h);
    k_mlp2<<<(B / 16) * (E / 16), 32, 0, stream>>>(h, W2, b2, mlp);
    k_softmax_e<<<B / 8, 256, 0, stream>>>(mlp, temp2, out);
}